// SmoothLoss_27822798143796
// MI455X (gfx1250) — compile-verified
//
#include <hip/hip_runtime.h>
#include <hip/hip_bf16.h>
#include <math.h>

typedef float v2f __attribute__((ext_vector_type(2)));
typedef float v8f __attribute__((ext_vector_type(8)));

#define NPTS 8192
#define DIM  128
#define NTIL (NPTS / 16)   // 512 16-point tiles

// ---------------------------------------------------------------------------
// Pad coords [N,3] -> [N,4] (w = 0 so K=4 WMMA gives the exact 3-D dot) and
// precompute squared norms.
// ---------------------------------------------------------------------------
__global__ void prep_kernel(const float* __restrict__ coords,
                            float* __restrict__ cpad,
                            float* __restrict__ sq) {
    int i = blockIdx.x * blockDim.x + threadIdx.x;
    if (i < NPTS) {
        float x = coords[3 * i + 0];
        float y = coords[3 * i + 1];
        float z = coords[3 * i + 2];
        cpad[4 * i + 0] = x;
        cpad[4 * i + 1] = y;
        cpad[4 * i + 2] = z;
        cpad[4 * i + 3] = 0.0f;
        sq[i] = x * x + y * y + z * z;
    }
}

// ---------------------------------------------------------------------------
// Brute-force nearest-neighbor via v_wmma_f32_16x16x4_f32.
// One wave32 owns one 16-row tile; loops over all 512 column tiles.
// A (16x4): lane l<16 holds A[M=l][K=0,1] in v0/v1; lane l+16 holds K=2,3.
// B (4x16): mirrored layout -> identical float2 load on the column tile.
// C/D: VGPR r <-> M = r + 8*(lane/16), N = lane%16.
// d2(m,n) = sq[m] + sq[n] - 2*dot(m,n); self excluded; running argmin per row.
// ---------------------------------------------------------------------------
__global__ void knn_kernel(const float* __restrict__ cpad,
                           const float* __restrict__ sq,
                           int* __restrict__ knn) {
    const int gtid = blockIdx.x * blockDim.x + threadIdx.x;
    const int wave = gtid >> 5;          // one wave per 16-row tile
    const int lane = threadIdx.x & 31;
    const int hf   = lane >> 4;          // half-wave select (K/M split)
    const int l16  = lane & 15;
    const int rowbase = wave * 16;

    // A tile load (row tile of coords, K padded to 4)
    v2f a;
    a.x = cpad[(rowbase + l16) * 4 + 2 * hf + 0];
    a.y = cpad[(rowbase + l16) * 4 + 2 * hf + 1];

    float sqm[8];
#pragma unroll
    for (int r = 0; r < 8; ++r) sqm[r] = sq[rowbase + r + 8 * hf];

    float best[8];
    int   bidx[8];
#pragma unroll
    for (int r = 0; r < 8; ++r) { best[r] = 3.4e38f; bidx[r] = 0; }

    for (int ct = 0; ct < NTIL; ++ct) {
        const int n = ct * 16 + l16;     // this lane's column index
        v2f b;
        b.x = cpad[n * 4 + 2 * hf + 0];
        b.y = cpad[n * 4 + 2 * hf + 1];
        const float sqn = sq[n];

        v8f c = {};
        // 8 args: (neg_a, A, neg_b, B, c_mod, C, reuse_a, reuse_b)
        c = __builtin_amdgcn_wmma_f32_16x16x4_f32(
                false, a, false, b, (short)0, c, false, false);

#pragma unroll
        for (int r = 0; r < 8; ++r) {
            const int m = rowbase + r + 8 * hf;
            float d2 = sqm[r] + sqn - 2.0f * c[r];
            if (n == m) d2 = 3.4e38f;            // exclude self
            if (d2 < best[r]) { best[r] = d2; bidx[r] = n; }
        }
    }

    // Reduce (min,idx) across the 16 lanes that share each row.
    // xor masks 8..1 never cross the 16-lane half-wave boundary.
#pragma unroll
    for (int r = 0; r < 8; ++r) {
        float bv = best[r];
        int   bi = bidx[r];
#pragma unroll
        for (int off = 8; off >= 1; off >>= 1) {
            float ov = __shfl_xor(bv, off);
            int   oi = __shfl_xor(bi, off);
            if (ov < bv || (ov == bv && oi < bi)) { bv = ov; bi = oi; }
        }
        if (l16 == 0) knn[rowbase + r + 8 * hf] = bi;
    }
}

// ---------------------------------------------------------------------------
// Fused L2 + TV partial sums. One wave per row (float4 per lane = 128 floats),
// 8 rows per block iteration. Per-block partials -> deterministic reduction.
// ---------------------------------------------------------------------------
__global__ void loss_kernel(const float* __restrict__ pred,
                            const float* __restrict__ target,
                            const int* __restrict__ knn,
                            float* __restrict__ partL2,
                            float* __restrict__ partTV) {
    const int lane = threadIdx.x & 31;
    const int wv   = threadIdx.x >> 5;            // 0..7
    const int rowsPerSweep = gridDim.x * 8;

    float l2 = 0.0f, tv = 0.0f;
    for (int row = blockIdx.x * 8 + wv; row < NPTS; row += rowsPerSweep) {
        const int j = knn[row];
        const float4 pv = ((const float4*)(pred   + (size_t)row * DIM))[lane];
        const float4 tg = ((const float4*)(target + (size_t)row * DIM))[lane];
        const float4 pj = ((const float4*)(pred   + (size_t)j   * DIM))[lane];
        float e;
        e = pv.x - tg.x; l2 += e * e;
        e = pv.y - tg.y; l2 += e * e;
        e = pv.z - tg.z; l2 += e * e;
        e = pv.w - tg.w; l2 += e * e;
        tv += fabsf(pv.x - pj.x) + fabsf(pv.y - pj.y) +
              fabsf(pv.z - pj.z) + fabsf(pv.w - pj.w);
    }

    __shared__ float sL[256];
    __shared__ float sT[256];
    sL[threadIdx.x] = l2;
    sT[threadIdx.x] = tv;
    __syncthreads();
    for (int s = 128; s > 0; s >>= 1) {
        if ((int)threadIdx.x < s) {
            sL[threadIdx.x] += sL[threadIdx.x + s];
            sT[threadIdx.x] += sT[threadIdx.x + s];
        }
        __syncthreads();
    }
    if (threadIdx.x == 0) {
        partL2[blockIdx.x] = sL[0];
        partTV[blockIdx.x] = sT[0];
    }
}

__global__ void finalize_kernel(const float* __restrict__ partL2,
                                const float* __restrict__ partTV,
                                float* __restrict__ out, int nb) {
    __shared__ float sL[256];
    __shared__ float sT[256];
    const int t = threadIdx.x;
    sL[t] = (t < nb) ? partL2[t] : 0.0f;
    sT[t] = (t < nb) ? partTV[t] : 0.0f;
    __syncthreads();
    for (int s = 128; s > 0; s >>= 1) {
        if (t < s) { sL[t] += sL[t + s]; sT[t] += sT[t + s]; }
        __syncthreads();
    }
    if (t == 0) {
        out[0] = sL[0] / (float)(NPTS * DIM) + sT[0] / (float)NPTS;
    }
}

// ---------------------------------------------------------------------------
extern "C" void kernel_launch(void* const* d_in, const int* in_sizes, int n_in,
                              void* d_out, int out_size, void* d_ws, size_t ws_size,
                              hipStream_t stream) {
    const float* pred   = (const float*)d_in[0];  // [8192,128] f32
    const float* target = (const float*)d_in[1];  // [8192,128] f32
    const float* coords = (const float*)d_in[2];  // [8192,3]   f32

    char* ws = (char*)d_ws;
    float* cpad = (float*)(ws);                     // 8192*4 f32 = 131072 B
    float* sq   = (float*)(ws + 131072);            // 8192   f32 =  32768 B
    int*   knn  = (int*)  (ws + 163840);            // 8192   i32 =  32768 B
    float* pL2  = (float*)(ws + 196608);            // 256    f32
    float* pTV  = (float*)(ws + 197632);            // 256    f32
    float* out  = (float*)d_out;

    prep_kernel<<<NPTS / 256, 256, 0, stream>>>(coords, cpad, sq);
    // 512 row tiles, 8 waves (256 threads) per block -> 64 blocks
    knn_kernel<<<NTIL / 8, 256, 0, stream>>>(cpad, sq, knn);
    loss_kernel<<<256, 256, 0, stream>>>(pred, target, knn, pL2, pTV);
    finalize_kernel<<<1, 256, 0, stream>>>(pL2, pTV, out, 256);
}